// LocalGraphProjection_2465311228492
// MI455X (gfx1250) — compile-verified
//
#include <hip/hip_runtime.h>
#include <math.h>

// ---------------------------------------------------------------------------
// LocalGraphProjection for MI455X (gfx1250, wave32)
//
// Roofline: output = 262144*339*4B ~= 355 MB of streaming stores -> ~15us at
// 23.3 TB/s HBM. Feature pyramids total 16.8 MB -> fully L2 (192MB) resident,
// so bilinear gathers are L2 traffic, not HBM. Kernel is store-bound:
//   - channel-parallel lanes => coalesced gathers and coalesced output stores
//   - non-temporal stores so the 355MB stream doesn't evict the feature maps
//   - point transform batched through V_WMMA_F32_16X16X4_F32 (16 points, all
//     3 views, in one matrix op): A = 9x4 affine rows, B = 4x16 points.
// ---------------------------------------------------------------------------

typedef __attribute__((ext_vector_type(2))) float v2f;
typedef __attribute__((ext_vector_type(8))) float v8f;

#define NPOINTS 262144

__device__ __forceinline__ float nan2num(float v) {
  if (__builtin_isnan(v)) return 0.0f;
  if (__builtin_isinf(v)) return v > 0.0f ? 3.402823466e38f : -3.402823466e38f;
  return v;
}

struct Corner {
  int o11, o21, o12, o22;
  float w11, w21, w12, w22;
};

// Bilinear corner offsets/weights for a (s, s, C) feature map, indices [x][y].
__device__ __forceinline__ Corner mk_corner(float h, float w, float invscale,
                                            int s, int C) {
  float x = fminf(fmaxf(h * invscale, 0.0f), (float)(s - 1));
  float y = fminf(fmaxf(w * invscale, 0.0f), (float)(s - 1));
  float x1 = floorf(x), x2 = ceilf(x);
  float y1 = floorf(y), y2 = ceilf(y);
  int x1i = (int)x1, x2i = (int)x2, y1i = (int)y1, y2i = (int)y2;
  Corner r;
  r.o11 = (x1i * s + y1i) * C;
  r.o21 = (x2i * s + y1i) * C;
  r.o12 = (x1i * s + y2i) * C;
  r.o22 = (x2i * s + y2i) * C;
  float dx2 = x2 - x, dx1 = x - x1, dy2 = y2 - y, dy1 = y - y1;
  r.w11 = dx2 * dy2;
  r.w21 = dx1 * dy2;
  r.w12 = dx2 * dy1;
  r.w22 = dx1 * dy1;
  return r;
}

__device__ __forceinline__ float bsample(const float* __restrict__ f,
                                         const Corner& c, int cc) {
  return c.w11 * f[c.o11 + cc] + c.w21 * f[c.o21 + cc] +
         c.w12 * f[c.o12 + cc] + c.w22 * f[c.o22 + cc];
}

// ---------------------------------------------------------------------------
// Prep kernel (1 thread): build the combined affine transforms.
//   point_origin = x @ inv(c0).T + o0
//   p_v          = (point_origin - o_v) @ c_v.T
//                = x @ M_v + t_v,   M_v[k][d] = sum_j inv(c0)[j][k] * c_v[d][j]
//                                   t_v[d]    = sum_j (o0 - o_v)[j] * c_v[d][j]
// Packed as a 16x4 A-matrix (rows r = 3v+d; col 3 = translation; rows 9..15=0)
// ---------------------------------------------------------------------------
__global__ void lgp_prep_kernel(const float* __restrict__ cam,
                                float* __restrict__ wsA) {
  if (threadIdx.x != 0) return;
  const float PI = 3.14159265358979323846f;
  float c[3][3][3];  // c[view][row][col]
  float o[3][3];
  for (int i = 0; i < 3; ++i) {
    const float* p = cam + i * 5;
    float theta = p[0] * PI / 180.0f;
    float phi = p[1] * PI / 180.0f;
    float camy = p[3] * sinf(phi);
    float lens = p[3] * cosf(phi);
    float camx = lens * cosf(theta);
    float camz = lens * sinf(theta);
    float Zv[3] = {camx, camy, camz};
    float Yv[3] = {camy * cosf(theta + PI), lens, camy * sinf(theta + PI)};
    float Xv[3] = {Yv[1] * Zv[2] - Yv[2] * Zv[1],
                   Yv[2] * Zv[0] - Yv[0] * Zv[2],
                   Yv[0] * Zv[1] - Yv[1] * Zv[0]};
    float nX = sqrtf(Xv[0] * Xv[0] + Xv[1] * Xv[1] + Xv[2] * Xv[2]);
    float nY = sqrtf(Yv[0] * Yv[0] + Yv[1] * Yv[1] + Yv[2] * Yv[2]);
    float nZ = sqrtf(Zv[0] * Zv[0] + Zv[1] * Zv[1] + Zv[2] * Zv[2]);
    for (int j = 0; j < 3; ++j) {
      c[i][0][j] = Xv[j] / nX;
      c[i][1][j] = Yv[j] / nY;
      c[i][2][j] = Zv[j] / nZ;
      o[i][j] = Zv[j];
    }
  }
  // N = inv(c[0]) via adjugate
  float a00 = c[0][0][0], a01 = c[0][0][1], a02 = c[0][0][2];
  float a10 = c[0][1][0], a11 = c[0][1][1], a12 = c[0][1][2];
  float a20 = c[0][2][0], a21 = c[0][2][1], a22 = c[0][2][2];
  float det = a00 * (a11 * a22 - a12 * a21) - a01 * (a10 * a22 - a12 * a20) +
              a02 * (a10 * a21 - a11 * a20);
  float id = 1.0f / det;
  float N[3][3];
  N[0][0] = (a11 * a22 - a12 * a21) * id;
  N[0][1] = (a02 * a21 - a01 * a22) * id;
  N[0][2] = (a01 * a12 - a02 * a11) * id;
  N[1][0] = (a12 * a20 - a10 * a22) * id;
  N[1][1] = (a00 * a22 - a02 * a20) * id;
  N[1][2] = (a02 * a10 - a00 * a12) * id;
  N[2][0] = (a10 * a21 - a11 * a20) * id;
  N[2][1] = (a01 * a20 - a00 * a21) * id;
  N[2][2] = (a00 * a11 - a01 * a10) * id;

  for (int v = 0; v < 3; ++v) {
    for (int d = 0; d < 3; ++d) {
      int r = 3 * v + d;
      for (int k = 0; k < 3; ++k) {
        float s = 0.0f;
        for (int j = 0; j < 3; ++j) s += N[j][k] * c[v][d][j];
        wsA[r * 4 + k] = s;
      }
      float t = 0.0f;
      for (int j = 0; j < 3; ++j) t += (o[0][j] - o[v][j]) * c[v][d][j];
      wsA[r * 4 + 3] = t;
    }
  }
  for (int r = 9; r < 16; ++r)
    for (int k = 0; k < 4; ++k) wsA[r * 4 + k] = 0.0f;
}

// ---------------------------------------------------------------------------
// Main kernel: one wave32 owns 16 points.
//   1. One V_WMMA_F32_16X16X4_F32: D[r][p] = transformed coord r of point p
//      (r = 3*view + dim, 9 rows used). Operand builds are branchless so
//      EXEC stays all-1s straight through the WMMA (ISA 7.12 requirement).
//   2. For each of the 16 points: broadcast its 9 coords (D rows 0..7 live in
//      lane=p VGPR r; row 8 lives in lane=p+16 VGPR 0), then all 32 lanes
//      sample channels in parallel and emit coalesced non-temporal stores.
// ---------------------------------------------------------------------------
__global__ void lgp_main_kernel(const float* __restrict__ pts,
                                const float* __restrict__ f0,
                                const float* __restrict__ f1,
                                const float* __restrict__ f2,
                                const float* __restrict__ wsA,
                                float* __restrict__ out) {
  const int lane = threadIdx.x & 31;
  const int waveId = blockIdx.x * (blockDim.x >> 5) + (threadIdx.x >> 5);
  const int base = waveId * 16;

  // A operand: 16x4 f32, lanes 0-15 hold K={0,1}, lanes 16-31 hold K={2,3}
  const int am = lane & 15;
  const int kh = lane >> 4;  // 0 for lanes 0-15, 1 for lanes 16-31
  v2f A;
  A.x = wsA[am * 4 + 2 * kh + 0];
  A.y = wsA[am * 4 + 2 * kh + 1];

  // B operand: 4x16 f32 points [x,y,z,1]; VGPR0 = rows K0/K2, VGPR1 = K1/K3.
  // Branchless: VGPR0 loads pp[0] (x) or pp[2] (z) via computed offset;
  // VGPR1 loads pp[1] (in-bounds for every lane) and selects 1.0 for K=3.
  const float* pp = pts + (size_t)(base + am) * 3;
  float by_mem = pp[1];
  v2f B;
  B.x = pp[kh * 2];                  // x (lanes 0-15) / z (lanes 16-31)
  B.y = kh ? 1.0f : by_mem;          // y (lanes 0-15) / 1.0 (lanes 16-31)

  v8f C = {0.f, 0.f, 0.f, 0.f, 0.f, 0.f, 0.f, 0.f};
  v8f D = __builtin_amdgcn_wmma_f32_16x16x4_f32(
      /*neg_a=*/false, A, /*neg_b=*/false, B,
      /*c_mod=*/(short)0, C, /*reuse_a=*/false, /*reuse_b=*/false);

  for (int p = 0; p < 16; ++p) {
    float co[9];
#pragma unroll
    for (int r = 0; r < 8; ++r) co[r] = __shfl(D[r], p, 32);
    co[8] = __shfl(D[0], p + 16, 32);

    const size_t pg = (size_t)base + p;
    float q[3][4];  // [view][channel-slot]  slots: 0:L0(c=lane,lane<16)
                    //                              1:L1(c=lane) 2/3:L2(c=lane,+32)
#pragma unroll
    for (int v = 0; v < 3; ++v) {
      float X = co[3 * v + 0];
      float Y = co[3 * v + 1];
      float Z = co[3 * v + 2];
      float h = nan2num(248.0f * (Y / Z) + 112.0f);     // (-Y)/(-Z)
      float w = nan2num(248.0f * (X / (-Z)) + 112.0f);

      const float* fv0 = f0 + (size_t)v * (224 * 224 * 16);
      Corner c0 = mk_corner(h, w, 1.0f, 224, 16);
      q[v][0] = (lane < 16) ? bsample(fv0, c0, lane) : 0.0f;

      const float* fv1 = f1 + (size_t)v * (112 * 112 * 32);
      Corner c1 = mk_corner(h, w, 0.5f, 112, 32);
      q[v][1] = bsample(fv1, c1, lane);

      const float* fv2 = f2 + (size_t)v * (56 * 56 * 64);
      Corner c2 = mk_corner(h, w, 0.25f, 56, 64);
      q[v][2] = bsample(fv2, c2, lane);
      q[v][3] = bsample(fv2, c2, lane + 32);
    }

    float* ob = out + pg * 339;
    if (lane < 3) __builtin_nontemporal_store(pts[pg * 3 + lane], ob + lane);

    const int chan[4] = {lane, 16 + lane, 48 + lane, 80 + lane};
#pragma unroll
    for (int s = 0; s < 4; ++s) {
      float a = q[0][s], b = q[1][s], cq = q[2][s];
      float mx = fmaxf(a, fmaxf(b, cq));
      float mn = (a + b + cq) * (1.0f / 3.0f);
      float d0 = a - mn, d1 = b - mn, d2 = cq - mn;
      float sd = sqrtf((d0 * d0 + d1 * d1 + d2 * d2) * (1.0f / 3.0f));
      bool active = (s != 0) || (lane < 16);
      if (active) {
        __builtin_nontemporal_store(mx, ob + 3 + chan[s]);
        __builtin_nontemporal_store(mn, ob + 115 + chan[s]);
        __builtin_nontemporal_store(sd, ob + 227 + chan[s]);
      }
    }
  }
}

extern "C" void kernel_launch(void* const* d_in, const int* in_sizes, int n_in,
                              void* d_out, int out_size, void* d_ws,
                              size_t ws_size, hipStream_t stream) {
  const float* pts = (const float*)d_in[0];   // (262144, 3)
  const float* cam = (const float*)d_in[1];   // (3, 5)
  const float* f0 = (const float*)d_in[2];    // (3, 224, 224, 16)
  const float* f1 = (const float*)d_in[3];    // (3, 112, 112, 32)
  const float* f2 = (const float*)d_in[4];    // (3, 56, 56, 64)
  float* out = (float*)d_out;                 // (262144, 339)
  float* wsA = (float*)d_ws;                  // 64 floats: 16x4 A-matrix

  lgp_prep_kernel<<<1, 32, 0, stream>>>(cam, wsA);

  constexpr int WAVES = NPOINTS / 16;          // 16384 waves, 16 pts each
  constexpr int WAVES_PER_BLOCK = 8;           // 256 threads, wave32
  constexpr int BLOCKS = WAVES / WAVES_PER_BLOCK;  // 2048
  lgp_main_kernel<<<BLOCKS, 256, 0, stream>>>(pts, f0, f1, f2, wsA, out);
}